// MoogVCFRFModelCell_55035710931666
// MI455X (gfx1250) — compile-verified
//
#include <hip/hip_runtime.h>

typedef float    v2f  __attribute__((ext_vector_type(2)));
typedef float    v8f  __attribute__((ext_vector_type(8)));
typedef unsigned v2ui __attribute__((ext_vector_type(2)));

#define K_STEP   (1.0f / 44100.0f)
#define R_PARAM  0.5f
#define F_PARAM  1000.0f
#define MAX_ITER 50

#if defined(__has_builtin)
#if __has_builtin(__builtin_amdgcn_permlane16_swap)
#define HAVE_PLSWAP 1
#endif
#endif

// tanh(x) = 1 - 2/(exp(2x)+1); exp(2x) = exp2(2*log2(e)*x).
// Overflow-safe: x->+inf gives exp2->inf, rcp->0, result 1; x->-inf gives -1.
__device__ __forceinline__ float fast_tanh(float x) {
  float e = __builtin_amdgcn_exp2f(x * 2.8853900817779268f);
  return __builtin_fmaf(-2.0f, __builtin_amdgcn_rcpf(e + 1.0f), 1.0f);
}

__device__ __forceinline__ float bperm(int idx_bytes, float v) {
  return __int_as_float(__builtin_amdgcn_ds_bpermute(idx_bytes, __float_as_int(v)));
}

// Merge D-layout rows into B-layout: result = {a.lanes0-15 | b.lanes0-15}.
// Fast path: single v_permlane16_swap_b32 (swaps a[16:31] <-> b[0:15]).
__device__ __forceinline__ float merge_lo_lo(float a, float b, int idx_hi, bool hi) {
#ifdef HAVE_PLSWAP
  v2ui r = __builtin_amdgcn_permlane16_swap(__float_as_uint(a), __float_as_uint(b),
                                            false, false);
  return __uint_as_float(r[0]);  // {a.lo | b.lo}
#else
  float t = bperm(idx_hi, b);    // lanes 16-31 <- b.lanes 0-15
  return hi ? t : a;
#endif
}

// Extract hi half down: result.lanes0-15 = a.lanes16-31 (lanes 16-31 don't-care).
__device__ __forceinline__ float hi_to_lo(float a, int idx_lo) {
#ifdef HAVE_PLSWAP
  v2ui r = __builtin_amdgcn_permlane16_swap(__float_as_uint(a), __float_as_uint(a),
                                            false, false);
  return __uint_as_float(r[1]);  // {a.hi | a.hi}
#else
  return bperm(idx_lo, a);
#endif
}

// One wave handles 16 batch elements. State kept in WMMA B-operand layout
// (4x16 f32, K-major): lane l holds components {k0, k0+1} of element (l&15),
// k0 = (l>=16) ? 2 : 0.  A operand = Kmat in rows 0-3 (16x4, rows 4-15 zero).
// D rows 0-3 land in VGPRs 0-3 lanes 0-15; rows 2,3 are merged back up to
// lanes 16-31 (permlane16_swap or ds_bpermute) to return to B-layout.
__global__ __launch_bounds__(256) void moog_vcf_newton_kernel(
    const float* __restrict__ u_n, const float* __restrict__ x_n1,
    const float* __restrict__ u_n1,
    const float* __restrict__ gf_p, const float* __restrict__ gr_p,
    const float* __restrict__ gainp_p, const float* __restrict__ alpha_p,
    float* __restrict__ out, int nbatch) {
  const int lane = threadIdx.x & 31;
  const int wave = (int)((blockIdx.x * blockDim.x + threadIdx.x) >> 5);
  const int base = wave * 16;
  if (base >= nbatch) return;              // wave-uniform: EXEC stays all-1s for WMMA

  const int  m  = lane & 15;               // element column / matrix row index
  const bool hi = lane >= 16;
  const int  e  = base + m;                // batch element for this lane
  const int  ec = (e < nbatch) ? e : (nbatch - 1);  // clamped for loads
  const int  k0 = hi ? 2 : 0;

  // ---------------- uniform scalar setup (update_matrices) ----------------
  const float gf = gf_p[0], gr = gr_p[0], gain_p = gainp_p[0], alpha = alpha_p[0];
  const float x4 = x_n1[3];                // previous_state[0,3]
  const float up = u_n1[0];                // previous_input[0]
  const float w  = 2.0f * 3.14159265358979323846f * gf * F_PARAM;
  float a14;
  if (x4 == 0.0f) a14 = -w * 4.0f * gr * R_PARAM;
  else            a14 = -w * fast_tanh(4.0f * gr * R_PARAM * x4) / fast_tanh(x4);
  const float xb  = gr * R_PARAM * x4;
  const float thx = fast_tanh(xb);
  const float miu = (xb == 0.0f) ? 1.0f : (thx / xb);
  const float beta = miu * (1.0f - thx * thx) / (1.0f - thx * fast_tanh(up));
  const float c2 = 1.0f - 0.5f * alpha;    // (1 - alpha/2)
  const float ca = 0.5f * alpha;           // alpha/2
  const float kbm = K_STEP * w * beta * c2;  // k_b_mul[0]
  const float kba = K_STEP * w * beta * ca;  // k_b_al[0]

  // A[m][k] structure: diag = -w, subdiag = +w, A[0][3] = a14, else 0.
  auto Aval = [&](int mm, int kk) -> float {
    if (mm >= 4) return 0.0f;
    if (kk == mm) return -w;
    if (kk == mm - 1) return w;
    if (mm == 0 && kk == 3) return a14;
    return 0.0f;
  };
  const float v0 = Aval(m, k0), v1 = Aval(m, k0 + 1);
  v2f aK;  aK.x  = v0 * (K_STEP * ca); aK.y  = v1 * (K_STEP * ca);  // Kmat
  v2f aM;  aM.x  = v0 * (K_STEP * c2); aM.y  = v1 * (K_STEP * c2);  // k_A_mul

  const int idx_hi = m << 2;                    // lanes 16-31 fetch lanes 0-15 (same m)
  const int idx_lo = ((lane + 16) & 31) << 2;   // lanes 0-15 fetch lanes 16-31

  // ---------------- per-element loads (B-layout) ----------------
  const float2 xv = *(const float2*)(x_n1 + (size_t)ec * 4 + k0);
  float x0 = xv.x, x1 = xv.y;
  const float u  = u_n[ec];
  const float u1 = u_n1[ec];

  // ---------------- p_n = x + k_A_mul @ tanh(x) + (kbm*u + kba*u1)*e0 ------
  {
    v2f bt; bt.x = fast_tanh(x0); bt.y = fast_tanh(x1);
    v8f zero = {0.f, 0.f, 0.f, 0.f, 0.f, 0.f, 0.f, 0.f};
    v8f pd = __builtin_amdgcn_wmma_f32_16x16x4_f32(false, aM, false, bt,
                                                   (short)0, zero, false, false);
    float pB0 = merge_lo_lo(pd[0], pd[2], idx_hi, hi) + x0;
    float pB1 = merge_lo_lo(pd[1], pd[3], idx_hi, hi) + x1;
    pB0 += hi ? 0.0f : __builtin_fmaf(kbm, u, kba * u1);  // component 0 only
    // keep p in D-layout as loop-invariant WMMA C operand (rows 0-3, lanes 0-15)
    float pd2 = hi_to_lo(pB0, idx_lo);
    float pd3 = hi_to_lo(pB1, idx_lo);
    v8f pC = {pB0, pB1, pd2, pd3, 0.f, 0.f, 0.f, 0.f};
    v8f mone = {-1.f, -1.f, -1.f, -1.f, -1.f, -1.f, -1.f, -1.f};

    // ---------------- 50 Newton iterations ----------------
#pragma unroll 1
    for (int it = 0; it < MAX_ITER; ++it) {
      float th0 = fast_tanh(x0), th1 = fast_tanh(x1);
      float s0 = __builtin_fmaf(-th0, th0, 1.0f);
      float s1 = __builtin_fmaf(-th1, th1, 1.0f);
      v2f bt2; bt2.x = th0; bt2.y = th1;
      v2f bs;  bs.x  = s0;  bs.y  = s1;
      // g rows = Kmat@th + p ; J rows = Kmat@(1-th^2) - 1
      v8f gD = __builtin_amdgcn_wmma_f32_16x16x4_f32(false, aK, false, bt2,
                                                     (short)0, pC, false, false);
      v8f jD = __builtin_amdgcn_wmma_f32_16x16x4_f32(false, aK, false, bs,
                                                     (short)0, mone, false, false);
      float gB0 = merge_lo_lo(gD[0], gD[2], idx_hi, hi) - x0;
      float gB1 = merge_lo_lo(gD[1], gD[3], idx_hi, hi) - x1;
      float jB0 = merge_lo_lo(jD[0], jD[2], idx_hi, hi);
      float jB1 = merge_lo_lo(jD[1], jD[3], idx_hi, hi);
      x0 = __builtin_fmaf(-gB0, __builtin_amdgcn_rcpf(jB0), x0);
      x1 = __builtin_fmaf(-gB1, __builtin_amdgcn_rcpf(jB1), x1);
    }
  }

  // ---------------- outputs: y [0,B), x [B,5B), u [5B,6B) ----------------
  if (e < nbatch) {
    float* outy = out;
    float* outx = out + (size_t)nbatch;
    float* outu = out + (size_t)nbatch * 5;
    float2 xs; xs.x = x0; xs.y = x1;
    *(float2*)(outx + (size_t)e * 4 + k0) = xs;
    if (hi) outy[e] = gain_p * x1;   // x3 lives in hi-lane VGPR1
    else    outu[e] = u;             // passthrough u_n
  }
}

extern "C" void kernel_launch(void* const* d_in, const int* in_sizes, int n_in,
                              void* d_out, int out_size, void* d_ws, size_t ws_size,
                              hipStream_t stream) {
  (void)n_in; (void)out_size; (void)d_ws; (void)ws_size;
  const float* u_n   = (const float*)d_in[0];
  const float* x_n1  = (const float*)d_in[1];
  const float* u_n1  = (const float*)d_in[2];
  const float* gf    = (const float*)d_in[3];
  const float* gr    = (const float*)d_in[4];
  const float* gainp = (const float*)d_in[5];
  const float* alpha = (const float*)d_in[6];
  float* out = (float*)d_out;

  const int n = in_sizes[0];                       // batch size B
  const long long waves   = ((long long)n + 15) / 16;
  const long long threads = waves * 32;
  const int block = 256;
  const int grid  = (int)((threads + block - 1) / block);
  moog_vcf_newton_kernel<<<grid, block, 0, stream>>>(
      u_n, x_n1, u_n1, gf, gr, gainp, alpha, out, n);
}